// A3TGCN_RecurrentGCN_89008902243189
// MI455X (gfx1250) — compile-verified
//
#include <hip/hip_runtime.h>
#include <math.h>

// ---------------------------------------------------------------------------
// A3TGCN (periods=1, H0=0) fused pipeline for MI455X / gfx1250 (wave32, WMMA)
//
// Simplifications proven from the reference:
//   * H0 == 0  =>  R-gate dead, bottom half of L*_W dead, H = (1-Z)*tanh(...)
//   * softmax(attention[1]) == 1.0
//   * agg(x@W) == agg(x)@W  =>  aggregate 16 features once, GEMMs after.
// Pipeline per 16-node tile (one wave):
//   GEMM1: agg16x16 @ [Wz|Wh]16x64 (+bias)        -> 16 v_wmma_f32_16x16x4_f32
//   GEMM2: gz,gh 16x32 @ Lz/Lh top-32x32 (+bias)  -> 32 wmma
//   act:   Z=sigmoid, Ht=tanh, h=relu((1-Z)*Ht)   (elementwise on C-frags)
//   GEMM3: h 16x32 @ out_W 32x10(pad16) (+bias)   ->  8 wmma
//   softmax over 10 classes per row via LDS.
// ---------------------------------------------------------------------------

typedef float v2f __attribute__((ext_vector_type(2)));
typedef float v8f __attribute__((ext_vector_type(8)));

#define NFD 16
#define SFD 32
#define NCD 10

static __device__ __forceinline__ v8f wmma4(v2f a, v2f b, v8f c) {
  // V_WMMA_F32_16X16X4_F32 : D(16x16 f32) = A(16x4 f32) * B(4x16 f32) + C
  return __builtin_amdgcn_wmma_f32_16x16x4_f32(
      /*neg_a=*/false, a, /*neg_b=*/false, b,
      /*c_mod=*/(short)0, c, /*reuse_a=*/false, /*reuse_b=*/false);
}

// Wave-private LDS phase separator: stop compiler reordering and drain DS ops.
static __device__ __forceinline__ void lds_phase() {
  __builtin_amdgcn_wave_barrier();
  asm volatile("s_wait_dscnt 0x0" ::: "memory");
  __builtin_amdgcn_wave_barrier();
}

static __device__ __forceinline__ void atom_add_f32(float* p, float v) {
  __hip_atomic_fetch_add(p, v, __ATOMIC_RELAXED, __HIP_MEMORY_SCOPE_AGENT);
}

// ---------------------------------------------------------------------------
// Graph-aggregation kernels
// ---------------------------------------------------------------------------

__global__ void k_init_deg(float* __restrict__ deg, int n) {
  int i = blockIdx.x * blockDim.x + threadIdx.x;
  if (i < n) deg[i] = 1.0f;  // self-loop weight
}

__global__ void k_degree(const int* __restrict__ ei, const float* __restrict__ w,
                         float* __restrict__ deg, int E) {
  int e = blockIdx.x * blockDim.x + threadIdx.x;
  if (e < E) atom_add_f32(&deg[ei[E + e]], w[e]);  // col = ei[1][e]
}

__global__ void k_dinv_selfloop(const float* __restrict__ deg,
                                const float* __restrict__ x,
                                float* __restrict__ dinv,
                                float* __restrict__ agg, int n) {
  int i = blockIdx.x * blockDim.x + threadIdx.x;
  if (i >= n) return;
  float d  = deg[i];
  float di = (d > 0.0f) ? rsqrtf(d) : 0.0f;
  dinv[i]  = di;
  float s  = di * di;  // self-loop norm: dinv[i]*1*dinv[i]
#pragma unroll
  for (int f = 0; f < NFD; ++f) agg[i * NFD + f] = s * x[i * NFD + f];
}

__global__ void k_edge_agg(const int* __restrict__ ei, const float* __restrict__ w,
                           const float* __restrict__ dinv, const float* __restrict__ x,
                           float* __restrict__ agg, int E) {
  int tid = blockIdx.x * blockDim.x + threadIdx.x;  // E*16 threads
  int e = tid >> 4;
  int f = tid & 15;
  if (e >= E) return;
  int r = ei[e];
  int c = ei[E + e];
  float nrm = dinv[r] * w[e] * dinv[c];
  atom_add_f32(&agg[c * NFD + f], nrm * x[r * NFD + f]);
}

// ---------------------------------------------------------------------------
// Fused dense WMMA kernel: one wave per 16-node tile, 8 waves per block.
// ---------------------------------------------------------------------------

__global__ __launch_bounds__(256) void k_dense(
    const float* __restrict__ agg,
    const float* __restrict__ Wz, const float* __restrict__ bz,
    const float* __restrict__ Wh, const float* __restrict__ bh,
    const float* __restrict__ LzW, const float* __restrict__ Lzb,
    const float* __restrict__ LhW, const float* __restrict__ Lhb,
    const float* __restrict__ oW, const float* __restrict__ ob,
    float* __restrict__ out, int nTiles) {
  __shared__ float ldsbuf[8 * 16 * 64];  // 4KB per wave

  const int wave = threadIdx.x >> 5;
  const int lane = threadIdx.x & 31;
  const int tile = blockIdx.x * 8 + wave;
  if (tile >= nTiles) return;

  float* L = ldsbuf + wave * 16 * 64;

  const int m    = lane & 15;        // A-row / B-col / C-col for this lane
  const int half = lane >> 4;        // lane group (0: lanes 0-15, 1: 16-31)
  const int kh   = half * 2;         // K sub-offset inside a K=4 chunk
  const int base = tile * 16;

  // ---------------- GEMM1: G = aggTile(16x16) @ [Wz|Wh](16x64) + [bz|bh] ---
  v8f g[4];
#pragma unroll
  for (int nt = 0; nt < 4; ++nt) {
    int col = nt * 16 + m;
    float bias = (col < SFD) ? bz[col] : bh[col - SFD];
#pragma unroll
    for (int r = 0; r < 8; ++r) g[nt][r] = bias;
  }
#pragma unroll
  for (int kc = 0; kc < 4; ++kc) {
    int k0 = kc * 4;
    v2f a;
    a.x = agg[(base + m) * NFD + k0 + kh];
    a.y = agg[(base + m) * NFD + k0 + kh + 1];
#pragma unroll
    for (int nt = 0; nt < 4; ++nt) {
      int col = nt * 16 + m;
      const float* W = (col < SFD) ? Wz : Wh;
      int cc = col & (SFD - 1);
      v2f b;
      b.x = W[(k0 + kh) * SFD + cc];
      b.y = W[(k0 + kh + 1) * SFD + cc];
      g[nt] = wmma4(a, b, g[nt]);
    }
  }

  // Stage G (16x64) through wave-private LDS: C-layout -> row-major.
#pragma unroll
  for (int nt = 0; nt < 4; ++nt)
#pragma unroll
    for (int r = 0; r < 8; ++r)
      L[(r + 8 * half) * 64 + nt * 16 + m] = g[nt][r];
  lds_phase();

  // ---------------- GEMM2: Zpre = gz@LzW[0:32], Htpre = gh@LhW[0:32] -------
  v8f zf[2], tf[2];
#pragma unroll
  for (int nt = 0; nt < 2; ++nt) {
    float bZ = Lzb[nt * 16 + m];
    float bH = Lhb[nt * 16 + m];
#pragma unroll
    for (int r = 0; r < 8; ++r) { zf[nt][r] = bZ; tf[nt][r] = bH; }
  }
#pragma unroll
  for (int kc = 0; kc < 8; ++kc) {
    int k0 = kc * 4;
    v2f az, ah;
    az.x = L[m * 64 + k0 + kh];
    az.y = L[m * 64 + k0 + kh + 1];
    ah.x = L[m * 64 + SFD + k0 + kh];
    ah.y = L[m * 64 + SFD + k0 + kh + 1];
#pragma unroll
    for (int nt = 0; nt < 2; ++nt) {
      int col = nt * 16 + m;
      v2f b1, b2;
      b1.x = LzW[(k0 + kh) * SFD + col];
      b1.y = LzW[(k0 + kh + 1) * SFD + col];
      zf[nt] = wmma4(az, b1, zf[nt]);
      b2.x = LhW[(k0 + kh) * SFD + col];
      b2.y = LhW[(k0 + kh + 1) * SFD + col];
      tf[nt] = wmma4(ah, b2, tf[nt]);
    }
  }
  lds_phase();  // WAR: done reading G before overwriting with h

  // ------------- Activations on C-fragments, write h (16x32) to LDS -------
#pragma unroll
  for (int nt = 0; nt < 2; ++nt)
#pragma unroll
    for (int r = 0; r < 8; ++r) {
      float Z  = 1.0f / (1.0f + __expf(-zf[nt][r]));
      float Ht = tanhf(tf[nt][r]);
      float H  = (1.0f - Z) * Ht;     // Z*H0 + (1-Z)*Ht with H0=0
      float hv = fmaxf(H, 0.0f);      // relu (attention weight == 1.0)
      L[(r + 8 * half) * SFD + nt * 16 + m] = hv;
    }
  lds_phase();

  // ---------------- GEMM3: logits = h(16x32) @ out_W(32x10 pad16) + ob ----
  v8f lg;
  {
    float bias = (m < NCD) ? ob[m] : 0.0f;
#pragma unroll
    for (int r = 0; r < 8; ++r) lg[r] = bias;
  }
#pragma unroll
  for (int kc = 0; kc < 8; ++kc) {
    int k0 = kc * 4;
    v2f a, b;
    a.x = L[m * SFD + k0 + kh];
    a.y = L[m * SFD + k0 + kh + 1];
    b.x = (m < NCD) ? oW[(k0 + kh) * NCD + m] : 0.0f;
    b.y = (m < NCD) ? oW[(k0 + kh + 1) * NCD + m] : 0.0f;
    lg = wmma4(a, b, lg);
  }
  lds_phase();  // WAR: done reading h before overwriting with logits

  // ---------------- Softmax over 10 classes, per row via LDS --------------
#pragma unroll
  for (int r = 0; r < 8; ++r) L[(r + 8 * half) * 16 + m] = lg[r];
  lds_phase();

  if (lane < 16) {
    int row = lane;
    float v[NCD];
    float mx = -3.4e38f;
#pragma unroll
    for (int j = 0; j < NCD; ++j) { v[j] = L[row * 16 + j]; mx = fmaxf(mx, v[j]); }
    float s = 0.0f;
#pragma unroll
    for (int j = 0; j < NCD; ++j) { v[j] = __expf(v[j] - mx); s += v[j]; }
    float inv = 1.0f / s;
#pragma unroll
    for (int j = 0; j < NCD; ++j) out[(base + row) * NCD + j] = v[j] * inv;
  }
}

// ---------------------------------------------------------------------------
// Launch
// ---------------------------------------------------------------------------

extern "C" void kernel_launch(void* const* d_in, const int* in_sizes, int n_in,
                              void* d_out, int out_size, void* d_ws, size_t ws_size,
                              hipStream_t stream) {
  const float* x   = (const float*)d_in[0];
  const int*   ei  = (const int*)  d_in[1];
  const float* ew  = (const float*)d_in[2];
  const float* Wz  = (const float*)d_in[3];
  const float* bz  = (const float*)d_in[4];
  // d_in[5], d_in[6] : Wr, br  (dead: H0 == 0)
  const float* Wh  = (const float*)d_in[7];
  const float* bh  = (const float*)d_in[8];
  const float* LzW = (const float*)d_in[9];
  const float* Lzb = (const float*)d_in[10];
  // d_in[11], d_in[12] : Lr_W, Lr_b (dead)
  const float* LhW = (const float*)d_in[13];
  const float* Lhb = (const float*)d_in[14];
  const float* oW  = (const float*)d_in[15];
  const float* ob  = (const float*)d_in[16];
  // d_in[17] : attention (softmax of 1 element == 1.0, dead)
  float* out = (float*)d_out;

  const int N = in_sizes[0] / NFD;
  const int E = in_sizes[1] / 2;

  float* ws   = (float*)d_ws;
  float* deg  = ws;            // N
  float* dinv = ws + N;        // N
  float* agg  = ws + 2 * N;    // N*16

  const int TB = 256;
  k_init_deg<<<(N + TB - 1) / TB, TB, 0, stream>>>(deg, N);
  k_degree<<<(E + TB - 1) / TB, TB, 0, stream>>>(ei, ew, deg, E);
  k_dinv_selfloop<<<(N + TB - 1) / TB, TB, 0, stream>>>(deg, x, dinv, agg, N);
  {
    long long tot = (long long)E * NFD;
    int blocks = (int)((tot + TB - 1) / TB);
    k_edge_agg<<<blocks, TB, 0, stream>>>(ei, ew, dinv, x, agg, E);
  }
  const int nTiles = (N + 15) / 16;
  k_dense<<<(nTiles + 7) / 8, TB, 0, stream>>>(agg, Wz, bz, Wh, bh, LzW, Lzb,
                                               LhW, Lhb, oW, ob, out, nTiles);
}